// photoreceptor_REIKE_54692113547672
// MI455X (gfx1250) — compile-verified
//
#include <hip/hip_runtime.h>
#include <cstdint>

#ifndef __has_builtin
#define __has_builtin(x) 0
#endif

#if __has_builtin(__builtin_amdgcn_global_load_async_to_lds_b32) && \
    __has_builtin(__builtin_amdgcn_s_wait_asynccnt)
#define USE_ASYNC_LDS 1
#else
#define USE_ASYNC_LDS 0
#endif

namespace {

constexpr int   kB     = 8;
constexpr int   kT     = 1000;
constexpr int   kD     = 1600;
constexpr int   kBlock = 256;   // 8 wave32s per block
constexpr int   kChunk = 32;    // time steps staged per double-buffer half
constexpr float kDt    = 0.008f;

// Builtin signature (from hipcc diagnostic): (int AS1*, int AS3*, imm, imm)
typedef __attribute__((address_space(1))) int g_i32;
typedef __attribute__((address_space(3))) int l_i32;

// x > 0 on all paths used here: pow(x,y) = exp2(y * log2(x)) on raw hw trans ops.
__device__ __forceinline__ float fast_pow(float x, float y) {
  return __builtin_amdgcn_exp2f(y * __builtin_amdgcn_logf(x));
}

} // namespace

extern "C" __global__ void __launch_bounds__(kBlock)
reike_photoreceptor_kernel(const float* __restrict__ X,
                           const float* __restrict__ pSigma,
                           const float* __restrict__ pPhi,
                           const float* __restrict__ pEta,
                           const float* __restrict__ pBeta,
                           const float* __restrict__ pC2c,
                           const float* __restrict__ pChill,
                           const float* __restrict__ pCdark,
                           const float* __restrict__ pHcoef,
                           const float* __restrict__ pHaff,
                           const float* __restrict__ pGamma,
                           const float* __restrict__ pGdark,
                           float* __restrict__ out) {
  // Double-buffered per-thread X staging: [2][kChunk][kBlock] floats = 64 KB.
  __shared__ float xtile[2 * kChunk * kBlock];

  const int tid = threadIdx.x;
  const int j   = blockIdx.x * kBlock + tid;   // chain id, 0..B*D-1 (exact grid)
  const int b   = j / kD;
  const int d   = j - b * kD;

  // ---- uniform parameter rescaling (matches reference preamble) ----
  const float sigma  = pSigma[0] * 100.0f;
  const float phi    = pPhi[0]   * 100.0f;
  const float eta    = pEta[0]   * 1000.0f;
  const float beta   = pBeta[0]  * 10.0f;
  const float c2c    = pC2c[0];
  const float chill  = pChill[0];
  const float cdark  = pCdark[0];
  const float hcoef  = pHcoef[0] * 10.0f;
  const float haff   = pHaff[0];
  const float gamma  = pGamma[0] * 0.125f;   // gamma / TIME_BIN
  const float gdark0 = pGdark[0] * 10.0f;

  const float darkCurrent = fast_pow(gdark0, chill) * c2c * 0.5f;
  const float gdark  = fast_pow(2.0f * darkCurrent / c2c, 1.0f / chill);
  const float cur2ca = beta * cdark / darkCurrent;
  const float smax   = eta / phi * gdark * (1.0f + fast_pow(cdark / haff, hcoef));
  const float kco    = cur2ca * c2c * 0.5f;  // c_n = c + dt*(kco*g^chill - beta*c)
  const float inv_haff = 1.0f / haff;
  const float outscale = -0.5f * c2c;        // out = -(c2c * g^chill) / 2

  const size_t base  = (size_t)b * kT * kD + (size_t)d;  // &X[b, t, d] = X + base + t*D
  const int    steps = kT - 1;                           // 999 scan steps

  // Stage kChunk consecutive time-steps of this thread's chain into LDS column `tid`.
  // Global addresses are coalesced across the wave (consecutive d). Each thread only
  // ever touches its own column, so no workgroup barrier is needed — only ASYNCcnt.
  auto stage = [&](int t0, int bufsel) {
    const float* gsrc = X + base + (size_t)t0 * kD;
    float*       ldst = &xtile[bufsel * kChunk * kBlock + tid];
#pragma unroll
    for (int k = 0; k < kChunk; ++k) {
      if (t0 + k < steps) {  // uniform predicate (t0, k uniform)
#if USE_ASYNC_LDS
        __builtin_amdgcn_global_load_async_to_lds_b32(
            (g_i32*)(uintptr_t)(gsrc + (size_t)k * kD),
            (l_i32*)(uintptr_t)(ldst + k * kBlock),  // AS3 ptr = low 32 bits = LDS offset
            0, 0);
#else
        ldst[k * kBlock] = gsrc[(size_t)k * kD];
#endif
      }
    }
  };

  stage(0, 0);
#if USE_ASYNC_LDS
  __builtin_amdgcn_s_wait_asynccnt(0);
#endif

  // ---- initial state (x0 = X[b,0,d], already staged) ----
  const float x0 = xtile[tid];
  float r  = x0 * gamma / sigma;
  float pp = (eta + r) / phi;
  float cc = cdark;
  float s  = gdark * eta / phi;
  float g  = gdark;
  float gp = fast_pow(gdark, chill);  // g^cgmphill carried across steps

  // g[0] == 0 in the reference -> out = -(c2c * 0^chill)/2 == -0.0f
  out[base] = outscale * 0.0f;

  int buf = 0;
  for (int t0 = 0; t0 < steps; t0 += kChunk) {
    const int tn = t0 + kChunk;
    if (tn < steps) stage(tn, buf ^ 1);  // overlap next chunk's HBM fetch with compute

    const int    kmax = (steps - t0 < kChunk) ? (steps - t0) : kChunk;
    const float* lsrc = &xtile[buf * kChunk * kBlock + tid];

    for (int k = 0; k < kmax; ++k) {
      const float x   = lsrc[k * kBlock];                    // ds_load, off critical cycle
      const float r_n = r + kDt * (-sigma * r) + gamma * x;
      const float p_n = pp + kDt * (r + eta - phi * pp);
      const float c_n = cc + kDt * (kco * gp - beta * cc);   // uses g^chill from prev step
      const float q   = fast_pow(c_n * inv_haff, hcoef);
      const float s_n = smax * __builtin_amdgcn_rcpf(1.0f + q);
      const float g_n = g + kDt * (s - pp * g);              // uses *old* s, pp (as in scan)
      gp = fast_pow(g_n, chill);                             // feeds next c AND the output
      out[base + (size_t)(t0 + k + 1) * kD] = outscale * gp;
      r = r_n; pp = p_n; cc = c_n; s = s_n; g = g_n;
    }

#if USE_ASYNC_LDS
    __builtin_amdgcn_s_wait_asynccnt(0);  // next buffer fully landed in LDS
#endif
    buf ^= 1;
  }
}

extern "C" void kernel_launch(void* const* d_in, const int* in_sizes, int n_in,
                              void* d_out, int out_size, void* d_ws, size_t ws_size,
                              hipStream_t stream) {
  (void)in_sizes; (void)n_in; (void)out_size; (void)d_ws; (void)ws_size;
  const float* X      = (const float*)d_in[0];
  const float* pSigma = (const float*)d_in[1];
  const float* pPhi   = (const float*)d_in[2];
  const float* pEta   = (const float*)d_in[3];
  const float* pBeta  = (const float*)d_in[4];
  const float* pC2c   = (const float*)d_in[5];
  const float* pChill = (const float*)d_in[6];
  const float* pCdark = (const float*)d_in[7];
  const float* pHcoef = (const float*)d_in[8];
  const float* pHaff  = (const float*)d_in[9];
  const float* pGamma = (const float*)d_in[10];
  const float* pGdark = (const float*)d_in[11];

  const int nChains = kB * kD;               // 12800
  dim3 grid(nChains / kBlock);               // 50 blocks x 256 threads (8 waves)
  reike_photoreceptor_kernel<<<grid, dim3(kBlock), 0, stream>>>(
      X, pSigma, pPhi, pEta, pBeta, pC2c, pChill, pCdark, pHcoef, pHaff,
      pGamma, pGdark, (float*)d_out);
}